// GraphStack_88596585382211
// MI455X (gfx1250) — compile-verified
//
#include <hip/hip_runtime.h>

#define BATCH  8
#define NNODE  2048
#define FDIM   128
#define NLAYER 2
#define SLOPE  0.1f
#define KSPLIT 8
#define LDSW   132   // padded row stride (132 % 64 == 4 -> conflict-free)

typedef float v2f __attribute__((ext_vector_type(2)));
typedef float v8f __attribute__((ext_vector_type(8)));

// CDNA5 fp32 WMMA: D(16x16,f32) = A(16x4,f32) * B(4x16,f32) + C
__device__ __forceinline__ v8f wmma4(v2f a, v2f b, v8f c) {
  return __builtin_amdgcn_wmma_f32_16x16x4_f32(
      /*neg_a=*/false, a, /*neg_b=*/false, b,
      /*c_mod=*/(short)0, c, /*reuse_a=*/false, /*reuse_b=*/false);
}

__device__ __forceinline__ float lrelu(float v) { return v >= 0.0f ? v : SLOPE * v; }

// ---------------------------------------------------------------------------
// Kernel 1: w1 = lrelu(x@W3+b3), w2 = lrelu(x@W4+b4), diag = rowdot(w1,w2)
// grid = B*N/16 blocks, 256 threads (8 waves; wave w owns output cols [16w,16w+16))
// ---------------------------------------------------------------------------
__global__ __launch_bounds__(256) void k_front(
    const float* __restrict__ x,
    const float* __restrict__ W3l, const float* __restrict__ b3l,
    const float* __restrict__ W4l, const float* __restrict__ b4l,
    float* __restrict__ w1, float* __restrict__ w2, float* __restrict__ diag) {
  __shared__ float xs[16 * LDSW];
  __shared__ float dwave[8][16];
  const int tid = threadIdx.x;
  const int lane = tid & 31, wave = tid >> 5;
  const int half = lane >> 4, idx = lane & 15;
  const long row0 = (long)blockIdx.x * 16;

  for (int e = tid; e < 16 * FDIM; e += 256) {
    int r = e >> 7, c = e & 127;
    xs[r * LDSW + c] = x[(row0 + r) * FDIM + c];
  }
  __syncthreads();

  const int col0 = wave * 16;
  v8f acc1 = {}; v8f acc2 = {};
  #pragma unroll 4
  for (int k = 0; k < FDIM; k += 4) {
    const int kk = k + half * 2;
    v2f a;  a.x  = xs[idx * LDSW + kk];           a.y  = xs[idx * LDSW + kk + 1];
    v2f b1; b1.x = W3l[kk * FDIM + col0 + idx];   b1.y = W3l[(kk + 1) * FDIM + col0 + idx];
    v2f b2; b2.x = W4l[kk * FDIM + col0 + idx];   b2.y = W4l[(kk + 1) * FDIM + col0 + idx];
    acc1 = wmma4(a, b1, acc1);
    acc2 = wmma4(a, b2, acc2);
  }

  const float bias1 = b3l[col0 + idx];
  const float bias2 = b4l[col0 + idx];
  float dp[8];
  #pragma unroll
  for (int r = 0; r < 8; ++r) {
    const int i = r + half * 8;                 // local row within 16-row chunk
    const float v1 = lrelu(acc1[r] + bias1);
    const float v2 = lrelu(acc2[r] + bias2);
    const long g = (row0 + i) * FDIM + col0 + idx;
    w1[g] = v1; w2[g] = v2;
    dp[r] = v1 * v2;
  }
  // deterministic butterfly reduction over the 16 columns of this wave's tile
  #pragma unroll
  for (int r = 0; r < 8; ++r) {
    #pragma unroll
    for (int m = 1; m < 16; m <<= 1) dp[r] += __shfl_xor(dp[r], m, 32);
  }
  if (idx == 0) {
    #pragma unroll
    for (int r = 0; r < 8; ++r) dwave[wave][half * 8 + r] = dp[r];
  }
  __syncthreads();
  if (tid < 16) {
    float s = 0.0f;
    #pragma unroll
    for (int w = 0; w < 8; ++w) s += dwave[w][tid];   // fixed order -> deterministic
    diag[row0 + tid] = s;
  }
}

// ---------------------------------------------------------------------------
// Kernel 2: split-K partials of S[b] = w2[b]^T @ x[b]  (F x F, K = N)
// grid = (F/16, B, KSPLIT), 256 threads; wave w owns g-cols [16w,16w+16)
// ---------------------------------------------------------------------------
__global__ __launch_bounds__(256) void k_scatter(
    const float* __restrict__ w2, const float* __restrict__ x, float* __restrict__ Sp) {
  const int tid = threadIdx.x;
  const int lane = tid & 31, wave = tid >> 5;
  const int half = lane >> 4, idx = lane & 15;
  const int f0 = blockIdx.x * 16;
  const int b  = blockIdx.y;
  const int j0 = blockIdx.z * (NNODE / KSPLIT);
  const int g0 = wave * 16;
  const float* w2b = w2 + (long)b * NNODE * FDIM;
  const float* xb  = x  + (long)b * NNODE * FDIM;

  v8f acc = {};
  for (int k = 0; k < NNODE / KSPLIT; k += 4) {
    const int j = j0 + k + half * 2;
    v2f a;  a.x  = w2b[(long)j * FDIM + f0 + idx];  a.y  = w2b[(long)(j + 1) * FDIM + f0 + idx];
    v2f bb; bb.x = xb [(long)j * FDIM + g0 + idx];  bb.y = xb [(long)(j + 1) * FDIM + g0 + idx];
    acc = wmma4(a, bb, acc);
  }
  float* Sb = Sp + ((long)blockIdx.z * BATCH + b) * FDIM * FDIM;
  #pragma unroll
  for (int r = 0; r < 8; ++r) {
    const int f = f0 + r + half * 8;
    Sb[f * FDIM + g0 + idx] = acc[r];
  }
}

// Fixed-order reduction of the KSPLIT partial S buffers (deterministic).
__global__ __launch_bounds__(256) void k_reduceS(
    const float* __restrict__ Sp, float* __restrict__ S) {
  const long i = (long)blockIdx.x * 256 + threadIdx.x;   // over B*F*F
  float s = 0.0f;
  #pragma unroll
  for (int z = 0; z < KSPLIT; ++z) s += Sp[(long)z * BATCH * FDIM * FDIM + i];
  S[i] = s;
}

// ---------------------------------------------------------------------------
// Kernel 3: msg = (w1@S - diag*x)/(N-1); out = lrelu(msg@W5+b5) + x   (fused)
// grid = B*N/16 blocks, 256 threads
// ---------------------------------------------------------------------------
__global__ __launch_bounds__(256) void k_back(
    const float* __restrict__ x, const float* __restrict__ w1,
    const float* __restrict__ diag, const float* __restrict__ S,
    const float* __restrict__ W5l, const float* __restrict__ b5l,
    float* __restrict__ xo) {
  __shared__ float w1s[16 * LDSW];
  __shared__ float xsv[16 * LDSW];
  __shared__ float msgs[16 * LDSW];
  __shared__ float dg[16];
  const int tid = threadIdx.x;
  const int lane = tid & 31, wave = tid >> 5;
  const int half = lane >> 4, idx = lane & 15;
  const long row0 = (long)blockIdx.x * 16;
  const int  b    = (int)(row0 >> 11);    // row0 / NNODE (N = 2048)

  for (int e = tid; e < 16 * FDIM; e += 256) {
    int r = e >> 7, c = e & 127;
    w1s[r * LDSW + c] = w1[(row0 + r) * FDIM + c];
    xsv[r * LDSW + c] = x [(row0 + r) * FDIM + c];
  }
  if (tid < 16) dg[tid] = diag[row0 + tid];
  __syncthreads();

  const int col0 = wave * 16;
  const float* Sb = S + (long)b * FDIM * FDIM;

  // GEMM 1: msg-tile = w1_chunk @ S[b]
  v8f acc = {};
  for (int k = 0; k < FDIM; k += 4) {
    const int kk = k + half * 2;
    v2f a;  a.x  = w1s[idx * LDSW + kk];           a.y  = w1s[idx * LDSW + kk + 1];
    v2f bb; bb.x = Sb[kk * FDIM + col0 + idx];     bb.y = Sb[(kk + 1) * FDIM + col0 + idx];
    acc = wmma4(a, bb, acc);
  }
  const float inv = 1.0f / (float)(NNODE - 1);
  #pragma unroll
  for (int r = 0; r < 8; ++r) {
    const int i = r + half * 8;
    const int g = col0 + idx;
    msgs[i * LDSW + g] = (acc[r] - dg[i] * xsv[i * LDSW + g]) * inv;
  }
  __syncthreads();

  // GEMM 2: out-tile = lrelu(msg_chunk @ W5 + b5) + x
  v8f acc2 = {};
  for (int k = 0; k < FDIM; k += 4) {
    const int kk = k + half * 2;
    v2f a;  a.x  = msgs[idx * LDSW + kk];          a.y  = msgs[idx * LDSW + kk + 1];
    v2f bb; bb.x = W5l[kk * FDIM + col0 + idx];    bb.y = W5l[(kk + 1) * FDIM + col0 + idx];
    acc2 = wmma4(a, bb, acc2);
  }
  const float bias = b5l[col0 + idx];
  #pragma unroll
  for (int r = 0; r < 8; ++r) {
    const int i = r + half * 8;
    const int g = col0 + idx;
    xo[(row0 + i) * FDIM + g] = lrelu(acc2[r] + bias) + xsv[i * LDSW + g];
  }
}

// ---------------------------------------------------------------------------
extern "C" void kernel_launch(void* const* d_in, const int* in_sizes, int n_in,
                              void* d_out, int out_size, void* d_ws, size_t ws_size,
                              hipStream_t stream) {
  const float* x  = (const float*)d_in[0];
  const float* W3 = (const float*)d_in[1];
  const float* b3 = (const float*)d_in[2];
  const float* W4 = (const float*)d_in[3];
  const float* b4 = (const float*)d_in[4];
  const float* W5 = (const float*)d_in[5];
  const float* b5 = (const float*)d_in[6];
  float* out = (float*)d_out;

  const long BNF = (long)BATCH * NNODE * FDIM;
  float* w1   = (float*)d_ws;                              // B*N*F
  float* w2   = w1 + BNF;                                  // B*N*F
  float* diag = w2 + BNF;                                  // B*N
  float* Sp   = diag + (long)BATCH * NNODE;                // KSPLIT*B*F*F
  float* S    = Sp + (long)KSPLIT * BATCH * FDIM * FDIM;   // B*F*F
  float* xtmp = S + (long)BATCH * FDIM * FDIM;             // B*N*F

  const int nrow_blocks = (BATCH * NNODE) / 16;            // 1024
  const float* xcur = x;
  for (int l = 0; l < NLAYER; ++l) {
    const float* W3l = W3 + (long)l * FDIM * FDIM;
    const float* W4l = W4 + (long)l * FDIM * FDIM;
    const float* W5l = W5 + (long)l * FDIM * FDIM;
    k_front<<<nrow_blocks, 256, 0, stream>>>(xcur, W3l, b3 + l * FDIM,
                                             W4l, b4 + l * FDIM, w1, w2, diag);
    k_scatter<<<dim3(FDIM / 16, BATCH, KSPLIT), 256, 0, stream>>>(w2, xcur, Sp);
    k_reduceS<<<(BATCH * FDIM * FDIM) / 256, 256, 0, stream>>>(Sp, S);
    float* xnext = (l == NLAYER - 1) ? out : xtmp;
    k_back<<<nrow_blocks, 256, 0, stream>>>(xcur, w1, diag, S, W5l, b5 + l * FDIM, xnext);
    xcur = xnext;
  }
}